// Precomputer_61718680043829
// MI455X (gfx1250) — compile-verified
//
#include <hip/hip_runtime.h>

typedef __attribute__((ext_vector_type(2)))  float v2f;
typedef __attribute__((ext_vector_type(8)))  float v8f;

#define NY      16
#define NRAD    34
#define NGRID   1001
#define FDELTA  0.01f          // X_MAX / (N_GRID-1)
#define INV_DELTA 100.0f
#define RCUT_IN 4.5f           // R_CUT - CUTOFF_WIDTH
#define TWO_PI  6.283185307179586f   // pi / CUTOFF_WIDTH

// Coefficient matrix B[20][16]: real spherical harmonics (Y00 = 1 normalization,
// i.e. reference Y * sqrt(4*pi)) expressed in the monomial basis
// {1, x,y,z, x2,xy,xz,y2,yz,z2, x3,x2y,x2z,xy2,xyz,xz2,y3,y2z,yz2,z3}.
__device__ __constant__ float BCOEF[20 * 16] = {
/* 1  */ 1.f,0.f,0.f,0.f, 0.f,0.f,-1.11803399f,0.f, 0.f,0.f,0.f,0.f, 0.f,0.f,0.f,0.f,
/* x  */ 0.f,0.f,0.f,1.73205081f, 0.f,0.f,0.f,0.f, 0.f,0.f,0.f,0.f, 0.f,-1.62018517f,0.f,0.f,
/* y  */ 0.f,1.73205081f,0.f,0.f, 0.f,0.f,0.f,0.f, 0.f,0.f,0.f,-1.62018517f, 0.f,0.f,0.f,0.f,
/* z  */ 0.f,0.f,1.73205081f,0.f, 0.f,0.f,0.f,0.f, 0.f,0.f,0.f,0.f, -3.96862697f,0.f,0.f,0.f,
/* x2 */ 0.f,0.f,0.f,0.f, 0.f,0.f,0.f,0.f, 1.93649167f,0.f,0.f,0.f, 0.f,0.f,0.f,0.f,
/* xy */ 0.f,0.f,0.f,0.f, 3.87298335f,0.f,0.f,0.f, 0.f,0.f,0.f,0.f, 0.f,0.f,0.f,0.f,
/* xz */ 0.f,0.f,0.f,0.f, 0.f,0.f,0.f,3.87298335f, 0.f,0.f,0.f,0.f, 0.f,0.f,0.f,0.f,
/* y2 */ 0.f,0.f,0.f,0.f, 0.f,0.f,0.f,0.f, -1.93649167f,0.f,0.f,0.f, 0.f,0.f,0.f,0.f,
/* yz */ 0.f,0.f,0.f,0.f, 0.f,3.87298335f,0.f,0.f, 0.f,0.f,0.f,0.f, 0.f,0.f,0.f,0.f,
/* z2 */ 0.f,0.f,0.f,0.f, 0.f,0.f,3.35410197f,0.f, 0.f,0.f,0.f,0.f, 0.f,0.f,0.f,0.f,
/* x3 */ 0.f,0.f,0.f,0.f, 0.f,0.f,0.f,0.f, 0.f,0.f,0.f,0.f, 0.f,0.f,0.f,2.09165007f,
/* x2y*/ 0.f,0.f,0.f,0.f, 0.f,0.f,0.f,0.f, 0.f,6.27495020f,0.f,0.f, 0.f,0.f,0.f,0.f,
/* x2z*/ 0.f,0.f,0.f,0.f, 0.f,0.f,0.f,0.f, 0.f,0.f,0.f,0.f, 0.f,0.f,5.12347538f,0.f,
/* xy2*/ 0.f,0.f,0.f,0.f, 0.f,0.f,0.f,0.f, 0.f,0.f,0.f,0.f, 0.f,0.f,0.f,-6.27495020f,
/* xyz*/ 0.f,0.f,0.f,0.f, 0.f,0.f,0.f,0.f, 0.f,0.f,10.24695077f,0.f, 0.f,0.f,0.f,0.f,
/* xz2*/ 0.f,0.f,0.f,0.f, 0.f,0.f,0.f,0.f, 0.f,0.f,0.f,0.f, 0.f,8.10092587f,0.f,0.f,
/* y3 */ 0.f,0.f,0.f,0.f, 0.f,0.f,0.f,0.f, 0.f,-2.09165007f,0.f,0.f, 0.f,0.f,0.f,0.f,
/* y2z*/ 0.f,0.f,0.f,0.f, 0.f,0.f,0.f,0.f, 0.f,0.f,0.f,0.f, 0.f,0.f,-5.12347538f,0.f,
/* yz2*/ 0.f,0.f,0.f,0.f, 0.f,0.f,0.f,0.f, 0.f,0.f,0.f,8.10092587f, 0.f,0.f,0.f,0.f,
/* z3 */ 0.f,0.f,0.f,0.f, 0.f,0.f,0.f,0.f, 0.f,0.f,0.f,0.f, 6.61437828f,0.f,0.f,0.f
};

// Main kernel: 256 threads = 8 wave32; each wave owns 16 pairs (one WMMA M-tile),
// block owns 128 pairs. Y via 5x chained v_wmma_f32_16x16x4_f32 (monomials x
// const coeffs, K = 20 = 5*4), radial via LDS-staged Hermite spline with
// coalesced final store. All WMMA operands live in VGPRs (no arrays -> no
// PromoteAlloca LDS round-trips).
__global__ __launch_bounds__(256) void pair_precompute(
    const float* __restrict__ positions,
    const float* __restrict__ cells,
    const int*   __restrict__ cell_shifts,
    const int*   __restrict__ center_idx,
    const int*   __restrict__ neigh_idx,
    const int*   __restrict__ struct_pairs,
    const int*   __restrict__ center_sp,
    const int*   __restrict__ neigh_sp,
    const float* __restrict__ lengthscales,
    const float* __restrict__ sval,
    const float* __restrict__ sder,
    float* __restrict__ outY,
    float* __restrict__ outR)
{
    __shared__ float s_x[128];
    __shared__ float s_cut[128];
    __shared__ float s_rad[128 * NRAD];

    const int  t    = threadIdx.x;
    const int  wave = t >> 5;
    const int  lane = t & 31;
    const int  col  = lane & 15;          // pair-in-tile / Y-component column
    const bool hi   = lane >= 16;
    const int  blockBase = blockIdx.x * 128;
    const int  tileBase  = blockBase + wave * 16;
    const int  p = tileBase + col;        // both half-waves replicate pairs 0..15

    // ---- B operands: lanes 0-15 take rows {4c,4c+1}, lanes 16-31 rows {4c+2,4c+3}
    // of the constant coefficient matrix; rows advance 4 per chunk -> +64 floats.
    const float* __restrict__ bp = BCOEF + col + (hi ? 32 : 0);
    v2f b0, b1, b2, b3, b4;
    b0.x = bp[0*64]; b0.y = bp[0*64 + 16];
    b1.x = bp[1*64]; b1.y = bp[1*64 + 16];
    b2.x = bp[2*64]; b2.y = bp[2*64 + 16];
    b3.x = bp[3*64]; b3.y = bp[3*64 + 16];
    b4.x = bp[4*64]; b4.y = bp[4*64 + 16];

    // ---- gather pair geometry ----
    const int ci = center_idx[p];
    const int ni = neigh_idx[p];
    const int sp = struct_pairs[p];
    const float sh0 = (float)cell_shifts[3 * p + 0];
    const float sh1 = (float)cell_shifts[3 * p + 1];
    const float sh2 = (float)cell_shifts[3 * p + 2];
    const float* __restrict__ C = cells + 9 * sp;   // 500 cells -> cache resident
    const float vx = positions[3*ni+0] - positions[3*ci+0] + sh0*C[0] + sh1*C[3] + sh2*C[6];
    const float vy = positions[3*ni+1] - positions[3*ci+1] + sh0*C[1] + sh1*C[4] + sh2*C[7];
    const float vz = positions[3*ni+2] - positions[3*ci+2] + sh0*C[2] + sh1*C[5] + sh2*C[8];
    const float rsq = vx*vx + vy*vy + vz*vz;
    const float r   = sqrtf(rsq);
    const float inv = 1.0f / r;
    const float ux = vx * inv, uy = vy * inv, uz = vz * inv;

    // per-pair scalars for the radial phase (lanes 0..15 only; hi half identical)
    if (!hi) {
        const float len = 0.1f + expf(lengthscales[center_sp[p]])
                               + expf(lengthscales[neigh_sp[p]]);
        s_x[wave * 16 + col]   = r / len;
        s_cut[wave * 16 + col] = (r < RCUT_IN) ? 1.0f
                               : 1.0f + cosf((r - RCUT_IN) * TWO_PI);
    }

    // ---- monomial basis (degree <= 3) as named scalars (stay in VGPRs) ----
    const float m0 = 1.0f, m1 = ux, m2 = uy, m3 = uz;
    const float m4 = ux*ux, m5 = ux*uy, m6 = ux*uz;
    const float m7 = uy*uy, m8 = uy*uz, m9 = uz*uz;
    const float m10 = m4*ux, m11 = m4*uy, m12 = m4*uz;
    const float m13 = ux*m7, m14 = m5*uz, m15 = ux*m9;
    const float m16 = uy*m7, m17 = m7*uz, m18 = uy*m9, m19 = uz*m9;

    // A 16x4 f32 layout: lanes 0-15 hold K={0,1}, lanes 16-31 hold K={2,3} of the
    // same row M (both halves carry the same pair's monomials).
    v2f a0, a1, a2, a3, a4;
    a0.x = hi ? m2  : m0 ;  a0.y = hi ? m3  : m1 ;
    a1.x = hi ? m6  : m4 ;  a1.y = hi ? m7  : m5 ;
    a2.x = hi ? m10 : m8 ;  a2.y = hi ? m11 : m9 ;
    a3.x = hi ? m14 : m12;  a3.y = hi ? m15 : m13;
    a4.x = hi ? m18 : m16;  a4.y = hi ? m19 : m17;

    v8f acc = {0.f,0.f,0.f,0.f,0.f,0.f,0.f,0.f};
    acc = __builtin_amdgcn_wmma_f32_16x16x4_f32(false, a0, false, b0, (short)0, acc, false, false);
    acc = __builtin_amdgcn_wmma_f32_16x16x4_f32(false, a1, false, b1, (short)0, acc, false, false);
    acc = __builtin_amdgcn_wmma_f32_16x16x4_f32(false, a2, false, b2, (short)0, acc, false, false);
    acc = __builtin_amdgcn_wmma_f32_16x16x4_f32(false, a3, false, b3, (short)0, acc, false, false);
    acc = __builtin_amdgcn_wmma_f32_16x16x4_f32(false, a4, false, b4, (short)0, acc, false, false);

    // ---- store Y tile: C/D layout VGPR j -> M = j (+8 for hi lanes), N = col ----
    {
        float* yo = outY + (size_t)(tileBase + (hi ? 8 : 0)) * NY + col;
        #pragma unroll
        for (int j = 0; j < 8; ++j) yo[j * NY] = acc[j];
    }

    __syncthreads();

    // ---- phase 2: Hermite spline, 2 threads/pair x 17 channels into LDS ----
    {
        const int   pl   = t >> 1;
        const int   half = t & 1;
        const float x    = s_x[pl];
        const float mc   = (x < 10.f) ? s_cut[pl] : 0.f;  // cutoff * (x<10) mask
        const float cx   = (x < 10.f) ? x : 5.f;
        int idx = (int)floorf(cx * INV_DELTA);
        idx = (idx < 0) ? 0 : ((idx > NGRID - 2) ? NGRID - 2 : idx);
        const float tt = (cx - (float)idx * FDELTA) * INV_DELTA;
        const float t2 = tt * tt, t3 = t2 * tt;
        const float w00 = ( 2.f*t3 - 3.f*t2 + 1.f) * mc;
        const float w10 = ( t3 - 2.f*t2 + tt) * FDELTA * mc;
        const float w01 = (-2.f*t3 + 3.f*t2) * mc;
        const float w11 = ( t3 - t2) * FDELTA * mc;
        const float* __restrict__ y0 = sval + (size_t)idx * NRAD + half * 17;
        const float* __restrict__ d0 = sder + (size_t)idx * NRAD + half * 17;
        __builtin_prefetch(y0, 0, 0);   // global_prefetch_b8 on gathered spline rows
        __builtin_prefetch(d0, 0, 0);
        float* ro = s_rad + pl * NRAD + half * 17;
        #pragma unroll
        for (int c2 = 0; c2 < 17; ++c2) {
            ro[c2] = w00 * y0[c2] + w10 * d0[c2]
                   + w01 * y0[c2 + NRAD] + w11 * d0[c2 + NRAD];
        }
    }
    __syncthreads();

    // ---- phase 3: coalesced dword store of the 128x34 radial tile ----
    {
        float* __restrict__ dst = outR + (size_t)blockBase * NRAD;
        #pragma unroll
        for (int k = 0; k < 17; ++k) {         // 17*256 = 4352 = 128*34
            const int q = k * 256 + t;
            dst[q] = s_rad[q];
        }
    }
}

// Scalar tail kernel for pair counts not divisible by 128 (dead for N_PAIRS=2M).
__global__ void pair_tail(
    const float* __restrict__ positions, const float* __restrict__ cells,
    const int* __restrict__ cell_shifts, const int* __restrict__ center_idx,
    const int* __restrict__ neigh_idx,   const int* __restrict__ struct_pairs,
    const int* __restrict__ center_sp,   const int* __restrict__ neigh_sp,
    const float* __restrict__ lengthscales,
    const float* __restrict__ sval, const float* __restrict__ sder,
    float* __restrict__ outY, float* __restrict__ outR, int start, int n)
{
    const int p = start + blockIdx.x * blockDim.x + threadIdx.x;
    if (p >= n) return;
    const int ci = center_idx[p], ni = neigh_idx[p], sp = struct_pairs[p];
    const float sh0 = (float)cell_shifts[3*p+0], sh1 = (float)cell_shifts[3*p+1],
                sh2 = (float)cell_shifts[3*p+2];
    const float* C = cells + 9 * sp;
    const float vx = positions[3*ni+0]-positions[3*ci+0]+sh0*C[0]+sh1*C[3]+sh2*C[6];
    const float vy = positions[3*ni+1]-positions[3*ci+1]+sh0*C[1]+sh1*C[4]+sh2*C[7];
    const float vz = positions[3*ni+2]-positions[3*ci+2]+sh0*C[2]+sh1*C[5]+sh2*C[8];
    const float r = sqrtf(vx*vx+vy*vy+vz*vz), inv = 1.0f / r;
    const float ux = vx*inv, uy = vy*inv, uz = vz*inv;
    float m[20];
    m[0]=1.0f; m[1]=ux; m[2]=uy; m[3]=uz;
    m[4]=ux*ux; m[5]=ux*uy; m[6]=ux*uz; m[7]=uy*uy; m[8]=uy*uz; m[9]=uz*uz;
    m[10]=m[4]*ux; m[11]=m[4]*uy; m[12]=m[4]*uz; m[13]=ux*m[7]; m[14]=m[5]*uz;
    m[15]=ux*m[9]; m[16]=uy*m[7]; m[17]=m[7]*uz; m[18]=uy*m[9]; m[19]=uz*m[9];
    for (int c = 0; c < NY; ++c) {
        float a = 0.f;
        #pragma unroll
        for (int row = 0; row < 20; ++row) a += m[row] * BCOEF[row * 16 + c];
        outY[(size_t)p * NY + c] = a;
    }
    const float len = 0.1f + expf(lengthscales[center_sp[p]]) + expf(lengthscales[neigh_sp[p]]);
    const float x = r / len;
    const float mc = (x < 10.f) ? ((r < RCUT_IN) ? 1.0f : 1.0f + cosf((r - RCUT_IN) * TWO_PI)) : 0.f;
    const float cx = (x < 10.f) ? x : 5.f;
    int idx = (int)floorf(cx * INV_DELTA);
    idx = (idx < 0) ? 0 : ((idx > NGRID - 2) ? NGRID - 2 : idx);
    const float tt = (cx - (float)idx * FDELTA) * INV_DELTA;
    const float t2 = tt*tt, t3 = t2*tt;
    const float w00 = ( 2.f*t3 - 3.f*t2 + 1.f) * mc;
    const float w10 = ( t3 - 2.f*t2 + tt) * FDELTA * mc;
    const float w01 = (-2.f*t3 + 3.f*t2) * mc;
    const float w11 = ( t3 - t2) * FDELTA * mc;
    const float* y0 = sval + (size_t)idx * NRAD;
    const float* d0 = sder + (size_t)idx * NRAD;
    for (int c = 0; c < NRAD; ++c)
        outR[(size_t)p * NRAD + c] = w00*y0[c] + w10*d0[c] + w01*y0[c+NRAD] + w11*d0[c+NRAD];
}

extern "C" void kernel_launch(void* const* d_in, const int* in_sizes, int n_in,
                              void* d_out, int out_size, void* d_ws, size_t ws_size,
                              hipStream_t stream) {
    const float* positions    = (const float*)d_in[0];
    const float* cells        = (const float*)d_in[1];
    const int*   cell_shifts  = (const int*)  d_in[2];
    const int*   center_idx   = (const int*)  d_in[3];
    const int*   neigh_idx    = (const int*)  d_in[4];
    const int*   struct_pairs = (const int*)  d_in[5];
    const int*   center_sp    = (const int*)  d_in[6];
    const int*   neigh_sp     = (const int*)  d_in[7];
    const float* lengthscales = (const float*)d_in[8];
    const float* sval         = (const float*)d_in[9];
    const float* sder         = (const float*)d_in[10];

    const int n = in_sizes[3];              // N_PAIRS
    float* outY = (float*)d_out;            // (n,16) then (n,34), concatenated
    float* outR = outY + (size_t)n * NY;

    const int nblocks = n / 128;            // 2,000,000 / 128 = 15625, exact
    if (nblocks > 0) {
        pair_precompute<<<nblocks, 256, 0, stream>>>(
            positions, cells, cell_shifts, center_idx, neigh_idx, struct_pairs,
            center_sp, neigh_sp, lengthscales, sval, sder, outY, outR);
    }
    const int start = nblocks * 128;
    const int rem = n - start;
    if (rem > 0) {
        pair_tail<<<(rem + 255) / 256, 256, 0, stream>>>(
            positions, cells, cell_shifts, center_idx, neigh_idx, struct_pairs,
            center_sp, neigh_sp, lengthscales, sval, sder, outY, outR, start, n);
    }
}